// EdgeGatedGraphConv_74637941670350
// MI455X (gfx1250) — compile-verified
//
#include <hip/hip_runtime.h>

// EdgeGatedGraphConv for MI455X (gfx1250), wave32, f32 WMMA 16x16x4 path.
// d_out = [x_new (40000x128) | y_new (640000x128)] f32.
// The y-region of d_out doubles as storage for the pre-BN edge values m.

#define NN 40000
#define NE 640000
#define DD 128
#define LDW 130   // padded LDS stride for W (bank-conflict-free b64 reads)

typedef float v2f __attribute__((ext_vector_type(2)));
typedef float v8f __attribute__((ext_vector_type(8)));

__device__ __forceinline__ unsigned short f2bf(float x) {
  unsigned int u = __float_as_uint(x);
  u += 0x7FFFu + ((u >> 16) & 1u);           // round-to-nearest-even
  return (unsigned short)(u >> 16);
}
__device__ __forceinline__ float bf2f(unsigned short h) {
  return __uint_as_float(((unsigned int)h) << 16);
}
__device__ __forceinline__ float softplusf(float x) {
  return fmaxf(x, 0.f) + log1pf(expf(-fabsf(x)));
}

// ---------------------------------------------------------------------------
// K1: nodeE[n][0:512] = x[n] @ {Wsg,Wdg,Wdu,Wsu}.T + bias  (blockIdx.y picks W)
// ---------------------------------------------------------------------------
__global__ void k_node_gemm(const float* __restrict__ x,
                            const float* __restrict__ W0, const float* __restrict__ W1,
                            const float* __restrict__ W2, const float* __restrict__ W3,
                            const float* __restrict__ b0, const float* __restrict__ b1,
                            const float* __restrict__ b2, const float* __restrict__ b3,
                            float* __restrict__ nodeE) {
  __shared__ float Wl[DD * LDW];
  const int wsel = blockIdx.y;
  const float* Wg = (wsel == 0) ? W0 : (wsel == 1) ? W1 : (wsel == 2) ? W2 : W3;
  const float* bg = (wsel == 0) ? b0 : (wsel == 1) ? b1 : (wsel == 2) ? b2 : b3;
  for (int i = threadIdx.x; i < DD * DD; i += blockDim.x)
    Wl[(i >> 7) * LDW + (i & 127)] = Wg[i];
  __syncthreads();

  const int lane = threadIdx.x & 31, wave = threadIdx.x >> 5;
  const int half = lane >> 4, l16 = lane & 15;
  const int tile = blockIdx.x * 8 + wave;
  if (tile >= NN / 16) return;
  const int row0 = tile * 16;

  float biasr[8];
#pragma unroll
  for (int nt = 0; nt < 8; nt++) biasr[nt] = bg[nt * 16 + l16];

  v8f acc[8];
#pragma unroll
  for (int nt = 0; nt < 8; nt++) acc[nt] = (v8f){0, 0, 0, 0, 0, 0, 0, 0};

  const float* arow = x + (size_t)(row0 + l16) * DD + 2 * half;
  for (int kt = 0; kt < 32; kt++) {
    v2f a = *(const v2f*)(arow + kt * 4);
    const float* wp = Wl + l16 * LDW + kt * 4 + 2 * half;
#pragma unroll
    for (int nt = 0; nt < 8; nt++) {
      v2f b = *(const v2f*)(wp + nt * 16 * LDW);
      acc[nt] = __builtin_amdgcn_wmma_f32_16x16x4_f32(false, a, false, b,
                                                      (short)0, acc[nt], false, false);
    }
  }
#pragma unroll
  for (int nt = 0; nt < 8; nt++) {
    const int f = nt * 16 + l16;
#pragma unroll
    for (int v = 0; v < 8; v++) {
      const int m = v + 8 * half;
      nodeE[(size_t)(row0 + m) * 512 + wsel * 128 + f] = acc[nt][v] + biasr[nt];
    }
  }
}

// ---------------------------------------------------------------------------
// CSR build: histogram of col, exclusive scan, placement
// ---------------------------------------------------------------------------
__global__ void k_hist(const int* __restrict__ eidx, int* __restrict__ counts) {
  int e = blockIdx.x * blockDim.x + threadIdx.x;
  if (e < NE) atomicAdd(&counts[eidx[NE + e]], 1);
}

__global__ void k_scan(const int* __restrict__ counts, int* __restrict__ offs,
                       int* __restrict__ cursor) {
  __shared__ int sdata[1024];
  __shared__ int srun;
  const int t = threadIdx.x;
  if (t == 0) srun = 0;
  __syncthreads();
  for (int base = 0; base < NN; base += 1024) {
    const int i = base + t;
    const int c = (i < NN) ? counts[i] : 0;
    sdata[t] = c;
    __syncthreads();
    for (int d = 1; d < 1024; d <<= 1) {
      int v = (t >= d) ? sdata[t - d] : 0;
      __syncthreads();
      sdata[t] += v;
      __syncthreads();
    }
    const int run = srun;
    if (i < NN) {
      const int ex = run + sdata[t] - c;
      offs[i] = ex;
      cursor[i] = ex;
    }
    __syncthreads();
    if (t == 0) srun = run + sdata[1023];
    __syncthreads();
  }
  if (t == 0) offs[NN] = srun;
}

__global__ void k_place(const int* __restrict__ eidx, int* __restrict__ cursor,
                        int* __restrict__ elist) {
  int e = blockIdx.x * blockDim.x + threadIdx.x;
  if (e < NE) {
    int pos = atomicAdd(&cursor[eidx[NE + e]], 1);
    elist[pos] = e;
  }
}

// ---------------------------------------------------------------------------
// K5: m = eattr @ We.T + be + Esrc[row] + Edst[col]; write m (d_out y-region),
//     sigma=sigmoid(m) as bf16, accumulate edge-BN sum/sumsq.
// ---------------------------------------------------------------------------
__global__ void k_edge_gemm(const float* __restrict__ eattr, const int* __restrict__ eidx,
                            const float* __restrict__ nodeE, const float* __restrict__ Wg,
                            const float* __restrict__ bg, float* __restrict__ yout,
                            unsigned short* __restrict__ sig, float* __restrict__ bnsums) {
  __shared__ float Wl[DD * LDW];
  __shared__ float bsum[128];
  __shared__ float bsq[128];
  for (int i = threadIdx.x; i < DD * DD; i += blockDim.x)
    Wl[(i >> 7) * LDW + (i & 127)] = Wg[i];
  if (threadIdx.x < 128) { bsum[threadIdx.x] = 0.f; bsq[threadIdx.x] = 0.f; }
  __syncthreads();

  const int lane = threadIdx.x & 31, wave = threadIdx.x >> 5;
  const int half = lane >> 4, l16 = lane & 15;

  float biasr[8], lsum[8], lsq[8];
#pragma unroll
  for (int nt = 0; nt < 8; nt++) { biasr[nt] = bg[nt * 16 + l16]; lsum[nt] = 0.f; lsq[nt] = 0.f; }

  const int nwaves = gridDim.x * 8;
  for (int tile = blockIdx.x * 8 + wave; tile < NE / 16; tile += nwaves) {
    const int e0 = tile * 16;
    v8f acc[8];
#pragma unroll
    for (int nt = 0; nt < 8; nt++) acc[nt] = (v8f){0, 0, 0, 0, 0, 0, 0, 0};

    const float* arow = eattr + (size_t)(e0 + l16) * DD + 2 * half;
    for (int kt = 0; kt < 32; kt++) {
      v2f a = *(const v2f*)(arow + kt * 4);
      const float* wp = Wl + l16 * LDW + kt * 4 + 2 * half;
#pragma unroll
      for (int nt = 0; nt < 8; nt++) {
        v2f b = *(const v2f*)(wp + nt * 16 * LDW);
        acc[nt] = __builtin_amdgcn_wmma_f32_16x16x4_f32(false, a, false, b,
                                                        (short)0, acc[nt], false, false);
      }
    }
#pragma unroll
    for (int v = 0; v < 8; v++) {
      const int e = e0 + v + 8 * half;
      const int re = eidx[e];
      const int ce = eidx[NE + e];
      const float* es = nodeE + (size_t)re * 512;        // Esrc
      const float* ed = nodeE + (size_t)ce * 512 + 128;  // Edst
#pragma unroll
      for (int nt = 0; nt < 8; nt++) {
        const int f = nt * 16 + l16;
        const float mval = acc[nt][v] + biasr[nt] + es[f] + ed[f];
        yout[(size_t)e * 128 + f] = mval;
        sig[(size_t)e * 128 + f] = f2bf(1.0f / (1.0f + __expf(-mval)));
        lsum[nt] += mval;
        lsq[nt] += mval * mval;
      }
    }
  }
#pragma unroll
  for (int nt = 0; nt < 8; nt++) {
    float s = lsum[nt] + __shfl_xor(lsum[nt], 16, 32);
    float q = lsq[nt] + __shfl_xor(lsq[nt], 16, 32);
    if (half == 0) {
      atomicAdd(&bsum[nt * 16 + l16], s);
      atomicAdd(&bsq[nt * 16 + l16], q);
    }
  }
  __syncthreads();
  if (threadIdx.x < 128) {
    atomicAdd(&bnsums[threadIdx.x], bsum[threadIdx.x]);
    atomicAdd(&bnsums[128 + threadIdx.x], bsq[threadIdx.x]);
  }
}

// ---------------------------------------------------------------------------
// K6: per node: s=sum sigma over incoming edges; z = U + Hdst*s/(s+eps);
//     write z (d_out x-region), accumulate node-BN sums.
// ---------------------------------------------------------------------------
__global__ void k_node_combine(const int* __restrict__ offs, const int* __restrict__ elist,
                               const unsigned short* __restrict__ sig,
                               const float* __restrict__ nodeE, float* __restrict__ xout,
                               float* __restrict__ bnsums) {
  const int f = threadIdx.x;  // 0..127
  float asum = 0.f, asq = 0.f;
  const int n0 = blockIdx.x * 32;
  for (int j = 0; j < 32; j++) {
    const int n = n0 + j;
    const int st = offs[n], en = offs[n + 1];
    float s = 0.f;
    for (int p = st; p < en; p++) {
      const int e = elist[p];
      s += bf2f(sig[(size_t)e * 128 + f]);
    }
    const float ratio = s / (s + 1e-6f);
    const float z = nodeE[(size_t)n * 512 + 384 + f] + nodeE[(size_t)n * 512 + 256 + f] * ratio;
    xout[(size_t)n * 128 + f] = z;
    asum += z;
    asq += z * z;
  }
  atomicAdd(&bnsums[256 + f], asum);
  atomicAdd(&bnsums[384 + f], asq);
}

// K7: finalize BN mean / rstd for nodes and edges
__global__ void k_bn_final(float* __restrict__ bn) {
  const int f = threadIdx.x;
  if (f < 128) {
    float em = bn[f] / (float)NE;
    float ev = fmaxf(bn[128 + f] / (float)NE - em * em, 0.f);
    float nm = bn[256 + f] / (float)NN;
    float nv = fmaxf(bn[384 + f] / (float)NN - nm * nm, 0.f);
    bn[512 + f] = nm;
    bn[640 + f] = rsqrtf(nv + 1e-5f);
    bn[768 + f] = em;
    bn[896 + f] = rsqrtf(ev + 1e-5f);
  }
}

// K8/K9: out = residual + softplus(gamma*(z-mean)*rstd + beta), float4, in place
__global__ void k_apply_out(const float* __restrict__ res, float* __restrict__ io,
                            const float* __restrict__ gamma, const float* __restrict__ beta,
                            const float* __restrict__ bnp) {
  const size_t i = ((size_t)blockIdx.x * blockDim.x + threadIdx.x) * 4;
  const int f0 = (int)(i & (size_t)(DD - 1));
  float4 zv = *(const float4*)(io + i);
  float4 rv = *(const float4*)(res + i);
  float z[4] = {zv.x, zv.y, zv.z, zv.w};
  float r[4] = {rv.x, rv.y, rv.z, rv.w};
  float o[4];
#pragma unroll
  for (int c = 0; c < 4; c++) {
    const int f = f0 + c;
    const float bnv = gamma[f] * (z[c] - bnp[f]) * bnp[128 + f] + beta[f];
    o[c] = r[c] + softplusf(bnv);
  }
  *(float4*)(io + i) = make_float4(o[0], o[1], o[2], o[3]);
}

// ---------------------------------------------------------------------------
extern "C" void kernel_launch(void* const* d_in, const int* in_sizes, int n_in,
                              void* d_out, int out_size, void* d_ws, size_t ws_size,
                              hipStream_t stream) {
  (void)in_sizes; (void)n_in; (void)out_size; (void)ws_size;
  const float* x    = (const float*)d_in[0];
  const int*   eidx = (const int*)d_in[1];
  const float* eatt = (const float*)d_in[2];
  const float* Wsg = (const float*)d_in[3];  const float* bsg = (const float*)d_in[4];
  const float* Wdg = (const float*)d_in[5];  const float* bdg = (const float*)d_in[6];
  const float* Weg = (const float*)d_in[7];  const float* beg = (const float*)d_in[8];
  const float* Wsu = (const float*)d_in[9];  const float* bsu = (const float*)d_in[10];
  const float* Wdu = (const float*)d_in[11]; const float* bdu = (const float*)d_in[12];
  const float* gn  = (const float*)d_in[13]; const float* btn = (const float*)d_in[14];
  const float* ge  = (const float*)d_in[15]; const float* bte = (const float*)d_in[16];

  float* xout = (float*)d_out;                       // [NN,128]
  float* yout = (float*)d_out + (size_t)NN * DD;     // [NE,128]; holds m, then y_new

  // workspace carve (~249 MB)
  char* w = (char*)d_ws;
  float* nodeE = (float*)w;                 w += (size_t)NN * 512 * 4;  // Esrc|Edst|Hdst|U
  unsigned short* sig = (unsigned short*)w; w += (size_t)NE * 128 * 2;  // sigma bf16
  int* counts = (int*)w;                    w += (size_t)NN * 4;
  int* offs   = (int*)w;                    w += (size_t)(NN + 1) * 4;
  int* cursor = (int*)w;                    w += (size_t)NN * 4;
  int* elist  = (int*)w;                    w += (size_t)NE * 4;
  float* bn   = (float*)w;                  // 1024 floats: sums + params

  hipMemsetAsync(counts, 0, (size_t)NN * 4, stream);
  hipMemsetAsync(bn, 0, 1024 * 4, stream);

  k_node_gemm<<<dim3((NN / 16 + 7) / 8, 4), 256, 0, stream>>>(
      x, Wsg, Wdg, Wdu, Wsu, bsg, bdg, bdu, bsu, nodeE);
  k_hist<<<NE / 256, 256, 0, stream>>>(eidx, counts);
  k_scan<<<1, 1024, 0, stream>>>(counts, offs, cursor);
  k_place<<<NE / 256, 256, 0, stream>>>(eidx, cursor, elist);
  k_edge_gemm<<<1250, 256, 0, stream>>>(eatt, eidx, nodeE, Weg, beg, yout, sig, bn);
  k_node_combine<<<NN / 32, 128, 0, stream>>>(offs, elist, sig, nodeE, xout, bn);
  k_bn_final<<<1, 128, 0, stream>>>(bn);
  k_apply_out<<<NN * DD / 4 / 256, 256, 0, stream>>>(x, xout, gn, btn, bn + 512);
  k_apply_out<<<NE * DD / 4 / 256, 256, 0, stream>>>(eatt, yout, ge, bte, bn + 768);
}